// SigLSTM_83631603188203
// MI455X (gfx1250) — compile-verified
//
#include <hip/hip_runtime.h>

// ---------------------------------------------------------------------------
// Fused signature-LSTM scan for MI455X (gfx1250, wave32).
//
// One wave owns 16 batch rows; lane pair (m, m+16) owns batch b0+m.
// Per timestep, two 16-row WMMA tiles over a shared K=32 operand
//   K = [ h(8) | x(3) | pad(1) | sig[1:](20) ]:
//   tileA rows = [ i(low units), chat(low) | i(high units), chat(high) ]  (3 K-chunks)
//   tileB rows = [ o(low),      f(low)     | o(high),       f(high)    ]  (8 K-chunks)
// using V_WMMA_F32_16X16X4_F32 (exact f32). Row permutation of the weight
// matrices makes every lane receive exactly the 16 gate values of its own
// 4 hidden units -> no cross-lane exchange on the critical path. Accumulation
// chains are ordered sig/x first, h last, so only 2 WMMAs trail the previous
// step's activations. Activations use native v_tanh_f32 when available.
// Depth-2 signature state (L1:4, L2:16 per batch) lives in registers and is
// advanced with Chen's identity; each lane of a pair owns the L2 columns its
// half-wave feeds into the B operand (low: j in {0,1}; high: j in {2,3}).
// ---------------------------------------------------------------------------

typedef float v2f __attribute__((ext_vector_type(2)));
typedef float v8f __attribute__((ext_vector_type(8)));

#define WMMA_F32(A, B, C) \
    __builtin_amdgcn_wmma_f32_16x16x4_f32(false, (A), false, (B), (short)0, (C), false, false)

#if __has_builtin(__builtin_amdgcn_tanhf)
__device__ __forceinline__ float tanh_f(float x) {
    return __builtin_amdgcn_tanhf(x);                  // v_tanh_f32 (1 trans op)
}
__device__ __forceinline__ float sigmoid_f(float x) {
    return fmaf(0.5f, __builtin_amdgcn_tanhf(0.5f * x), 0.5f);
}
#else
__device__ __forceinline__ float sigmoid_f(float x) {
    return 1.0f / (1.0f + __expf(-x));
}
__device__ __forceinline__ float tanh_f(float x) {
    return fmaf(2.0f, sigmoid_f(2.0f * x), -1.0f);
}
#endif

#if __has_builtin(__builtin_amdgcn_rcpf)
__device__ __forceinline__ float rcp_f(float x) { return __builtin_amdgcn_rcpf(x); }
#else
__device__ __forceinline__ float rcp_f(float x) { return 1.0f / x; }
#endif

__global__ __launch_bounds__(32) void siglstm_scan_kernel(
    const float* __restrict__ inp,   // (B, T, 3)
    const float* __restrict__ ik,    // (3, 24)  input_kernel
    const float* __restrict__ rk,    // (8, 24)  recurrent_kernel
    const float* __restrict__ fk,    // (21, 8)  forget_kernel
    const float* __restrict__ bias,  // (32,)    [b_i, b_f, b_c, b_o]
    float* __restrict__ out,         // (B, T, 8)
    int T)
{
    const int  lane = threadIdx.x;          // 0..31
    const bool hi   = lane >= 16;
    const int  m    = lane & 15;
    const int  b    = blockIdx.x * 16 + m;  // batch row owned by this lane pair
    const int  klow = hi ? 2 : 0;           // K offset of this half-wave in A/B chunks
    const int  u0   = hi ? 2 : 0;           // first hidden unit owned by this lane

    // ----- row permutation ---------------------------------------------------
    const int rlow = m & 7;
    const int grp  = m >> 3;                // 0: rows for low lanes, 1: for high lanes
    const int slot = rlow & 3;
    const int kind = rlow >> 2;             // tileA: 0=i,1=chat ; tileB: 0=o,1=f
    const int uu   = (grp ? 2 : 0) + (slot & 1) + ((slot >> 1) << 2);  // unit of row m

    // W_cat[k][g]: k<8 -> R[k][g]; k<11 -> Wx[k-8][g]; else 0.
    auto wcat = [&](int k, int g) -> float {
        if (k < 8)  return rk[k * 24 + g];
        if (k < 11) return ik[(k - 8) * 24 + g];
        return 0.0f;
    };

    const int gA = kind ? (8 + uu) : uu;    // tileA row's gate (i: 0..7, chat: 8..15)
    const int gB = 16 + uu;                 // tileB o-row's gate

    v2f aA[3], aB[8];
#pragma unroll
    for (int c = 0; c < 3; ++c) {
        int k0 = 4 * c + klow;
        aA[c][0] = wcat(k0,     gA);
        aA[c][1] = wcat(k0 + 1, gA);
    }
#pragma unroll
    for (int c = 0; c < 8; ++c) {
        int k0 = 4 * c + klow;
        int k1 = k0 + 1;
        float b0, b1;
        if (kind == 0) {                    // o row: support in K 0..10
            b0 = wcat(k0, gB);
            b1 = wcat(k1, gB);
        } else {                            // f row: support in K 12..31, s = K-11
            b0 = (k0 >= 12) ? fk[(k0 - 11) * 8 + uu] : 0.0f;
            b1 = (k1 >= 12) ? fk[(k1 - 11) * 8 + uu] : 0.0f;
        }
        aB[c][0] = b0;
        aB[c][1] = b1;
    }

    // ----- C-matrix initializers (biases), in MY half's unit order ----------
    v8f cAv, cBv;
    float wfv[4];                           // Wf[0][u]: const-sig-term column
#pragma unroll
    for (int r = 0; r < 8; ++r) {
        int s  = r & 3;
        int ur = u0 + (s & 1) + ((s >> 1) << 2);
        if (r < 4) {
            cAv[r] = bias[ur];              // b_i
            cBv[r] = bias[24 + ur];         // b_o
        } else {
            cAv[r] = bias[16 + ur];         // b_c
            cBv[r] = bias[8 + ur];          // b_f
            wfv[r - 4] = fk[ur];
        }
    }

    // ----- per-batch state ---------------------------------------------------
    float L1[4]  = {0.f, 0.f, 0.f, 0.f};    // level-1 signature (full, both lanes)
    float L2a[4] = {0.f, 0.f, 0.f, 0.f};    // L2[i][j0], j0 = hi?2:0
    float L2b[4] = {0.f, 0.f, 0.f, 0.f};    // L2[i][j1], j1 = hi?3:1
    float hslot[4] = {0.f, 0.f, 0.f, 0.f};  // units {u0,u0+1,u0+4,u0+5}
    float cslot[4] = {0.f, 0.f, 0.f, 0.f};
    float invt = 1.0f;                      // sig[0]=1 at t=0 (no /t at t=0)
    const float dt = 1.0f / (float)(T - 1);

    const float* xrow = inp + (size_t)b * T * 3;
    float x0 = xrow[0], x1 = xrow[1], x2 = xrow[2];

    for (int t = 0; t < T; ++t) {
        // ----- B operands over K = [h | x | pad | sig] -----------------------
        v2f bh0, bh1, bx;
        bh0[0] = hslot[0]; bh0[1] = hslot[1];          // K {0,1}|{2,3}
        bh1[0] = hslot[2]; bh1[1] = hslot[3];          // K {4,5}|{6,7}
        bx[0]  = hi ? x2 : x0;                         // K {8,9}|{10,11(pad)}
        bx[1]  = hi ? 0.0f : x1;

        v2f bs3, bs4, bs5, bs6, bs7;
        bs3[0] = (hi ? L1[2] : L1[0]) * invt;          // s {1,2}|{3,4}
        bs3[1] = (hi ? L1[3] : L1[1]) * invt;
        bs4[0] = L2a[0] * invt; bs4[1] = L2b[0] * invt;
        bs5[0] = L2a[1] * invt; bs5[1] = L2b[1] * invt;
        bs6[0] = L2a[2] * invt; bs6[1] = L2b[2] * invt;
        bs7[0] = L2a[3] * invt; bs7[1] = L2b[3] * invt;

        // ----- matrix core: 11 x v_wmma_f32_16x16x4_f32 ----------------------
        // Accumulate x/sig contributions first; h-dependent chunks last, so
        // only 2 WMMAs trail the previous step's activations on each chain.
        v8f dA = cAv;                                  // rows: i | chat (my units)
        dA = WMMA_F32(aA[2], bx,  dA);
        dA = WMMA_F32(aA[1], bh1, dA);
        dA = WMMA_F32(aA[0], bh0, dA);

        v8f dB1;                                       // rows: o | f (my units)
#pragma unroll
        for (int r = 0; r < 8; ++r)
            dB1[r] = (r < 4) ? cBv[r] : fmaf(wfv[r - 4], invt, cBv[r]);
        dB1 = WMMA_F32(aB[3], bs3, dB1);
        dB1 = WMMA_F32(aB[2], bx,  dB1);
        dB1 = WMMA_F32(aB[1], bh1, dB1);
        dB1 = WMMA_F32(aB[0], bh0, dB1);

        v8f dB2 = {};                                  // sig-only chain (h-independent)
        dB2 = WMMA_F32(aB[4], bs4, dB2);
        dB2 = WMMA_F32(aB[5], bs5, dB2);
        dB2 = WMMA_F32(aB[6], bs6, dB2);
        dB2 = WMMA_F32(aB[7], bs7, dB2);

        v8f dB = dB1 + dB2;

        // ----- pointwise LSTM update (no cross-lane traffic) -----------------
#pragma unroll
        for (int s = 0; s < 4; ++s) {
            float ia = sigmoid_f(dA[s]);
            float ca = tanh_f(dA[4 + s]);
            float oa = sigmoid_f(dB[s]);
            float fa = sigmoid_f(dB[4 + s]);
            float cn = fmaf(fa, cslot[s], ia * ca);
            cslot[s] = cn;
            hslot[s] = oa * tanh_f(cn);
        }

        // ----- store h[b][t][u] for owned units (2 x 8-byte stores) ----------
        size_t ob = ((size_t)b * T + t) * 8 + u0;
        *reinterpret_cast<float2*>(out + ob)     = make_float2(hslot[0], hslot[1]);
        *reinterpret_cast<float2*>(out + ob + 4) = make_float2(hslot[2], hslot[3]);

        // ----- advance path / signature state (Chen's identity) --------------
        int tn = (t + 1 < T) ? (t + 1) : (T - 1);
        const float* xp = xrow + (size_t)tn * 3;
        float xn0 = xp[0], xn1 = xp[1], xn2 = xp[2];
        int tp = (t + 16 < T) ? (t + 16) : (T - 1);
        __builtin_prefetch(xrow + (size_t)tp * 3, 0, 1);  // global_prefetch_b8

        float dx0 = dt;
        float dx1 = xn0 - x0, dx2 = xn1 - x1, dx3 = xn2 - x2;
        float dxa = hi ? dx2 : dx0;                    // my j0 column increment
        float dxb = hi ? dx3 : dx1;                    // my j1 column increment
        float dxv[4] = {dx0, dx1, dx2, dx3};
#pragma unroll
        for (int i = 0; i < 4; ++i) {
            float tmp = fmaf(0.5f, dxv[i], L1[i]);     // L1_prev[i] + 0.5*dx[i]
            L2a[i] = fmaf(tmp, dxa, L2a[i]);
            L2b[i] = fmaf(tmp, dxb, L2b[i]);
        }
#pragma unroll
        for (int i = 0; i < 4; ++i) L1[i] += dxv[i];
        x0 = xn0; x1 = xn1; x2 = xn2;
        invt = (float)(T - 1) * rcp_f((float)(t + 1)); // t_grid normalization for t+1
    }
}

extern "C" void kernel_launch(void* const* d_in, const int* in_sizes, int n_in,
                              void* d_out, int out_size, void* d_ws, size_t ws_size,
                              hipStream_t stream) {
    (void)in_sizes; (void)n_in; (void)d_ws; (void)ws_size; (void)out_size;
    const float* inp  = (const float*)d_in[0];  // (256, 4096, 3)
    const float* ik   = (const float*)d_in[1];  // (3, 24)
    const float* rk   = (const float*)d_in[2];  // (8, 24)
    const float* fk   = (const float*)d_in[3];  // (21, 8)
    const float* bias = (const float*)d_in[4];  // (32,)
    float* out = (float*)d_out;                 // (256, 4096, 8)

    const int B = 256, T = 4096;
    dim3 grid(B / 16), block(32);               // 16 waves, one per WGP
    siglstm_scan_kernel<<<grid, block, 0, stream>>>(inp, ik, rk, fk, bias, out, T);
}